// EdgeAggragate_48627619726064
// MI455X (gfx1250) — compile-verified
//
#include <hip/hip_runtime.h>

typedef __attribute__((ext_vector_type(2))) float v2f;
typedef __attribute__((ext_vector_type(8))) float v8f;

#define E_CNT 500000
#define P_CNT 4000000
#define D_DIM 64
#define U_DIM 64

// ---------------------------------------------------------------------------
// Stage 1: T = features @ kernel    ([E,64] x [64,64] -> [E,64], fp32)
// One wave per 16-row tile. 4 accumulators (16x16 each) cover N = 64.
// Uses V_WMMA_F32_16X16X4_F32, 16 k-steps for K = 64.
//
// fp32 A 16x4 fragment layout (ISA 7.12.2): lane L (half = L>>4, m = L&15):
//   a.x = A[m, kbase + half*2],  a.y = A[m, kbase + half*2 + 1]
// fp32 B 4x16 fragment mirrors it with N = m striped across lanes.
// fp32 C/D 16x16: vgpr v holds row (v + half*8), col m.
// ---------------------------------------------------------------------------
__global__ __launch_bounds__(256) void gemm_wmma_kernel(
    const float* __restrict__ A,      // [E, 64] features
    const float* __restrict__ B,      // [64, 64] kernel
    float* __restrict__ T,            // [E, 64] transformed (workspace)
    int nRows)
{
    const int wave = blockIdx.x * (blockDim.x >> 5) + (threadIdx.x >> 5);
    const int r0 = wave * 16;
    if (r0 >= nRows) return;          // wave-uniform: EXEC stays all-ones

    const int lane = threadIdx.x & 31;
    const int half = lane >> 4;       // 0: K pair {0,1}, 1: K pair {2,3}
    const int m    = lane & 15;       // row (A) / col (B,C,D) within tile

    v8f acc0 = {}, acc1 = {}, acc2 = {}, acc3 = {};

    const float* __restrict__ arow = A + (size_t)(r0 + m) * D_DIM;

#pragma unroll
    for (int kk = 0; kk < 16; ++kk) {
        const int kbase = kk * 4 + half * 2;

        v2f a;
        a.x = arow[kbase];
        a.y = arow[kbase + 1];

        const float* __restrict__ b0 = B + (size_t)kbase * U_DIM + m;
        const float* __restrict__ b1 = b0 + U_DIM;

        v2f b;
        b.x = b0[0];  b.y = b1[0];
        acc0 = __builtin_amdgcn_wmma_f32_16x16x4_f32(false, a, false, b,
                                                     (short)0, acc0, false, false);
        b.x = b0[16]; b.y = b1[16];
        acc1 = __builtin_amdgcn_wmma_f32_16x16x4_f32(false, a, false, b,
                                                     (short)0, acc1, false, false);
        b.x = b0[32]; b.y = b1[32];
        acc2 = __builtin_amdgcn_wmma_f32_16x16x4_f32(false, a, false, b,
                                                     (short)0, acc2, false, false);
        b.x = b0[48]; b.y = b1[48];
        acc3 = __builtin_amdgcn_wmma_f32_16x16x4_f32(false, a, false, b,
                                                     (short)0, acc3, false, false);
    }

    // D layout: vgpr v -> row r0 + half*8 + v, col m (+ 16*nt)
#pragma unroll
    for (int v = 0; v < 8; ++v) {
        float* __restrict__ o = T + (size_t)(r0 + half * 8 + v) * U_DIM + m;
        o[0]  = acc0[v];
        o[16] = acc1[v];
        o[32] = acc2[v];
        o[48] = acc3[v];
    }
}

// ---------------------------------------------------------------------------
// Stage 2: out[e, :] = bias  (float4-vectorized 128 MB stream write)
// ---------------------------------------------------------------------------
__global__ __launch_bounds__(256) void bias_init_kernel(
    const float* __restrict__ bias,
    float4* __restrict__ out,
    int nQuads)
{
    const int i = blockIdx.x * blockDim.x + threadIdx.x;
    if (i >= nQuads) return;
    const int d = (i & (U_DIM / 4 - 1)) * 4;   // (i*4) & 63
    float4 b;
    b.x = bias[d + 0];
    b.y = bias[d + 1];
    b.z = bias[d + 2];
    b.w = bias[d + 3];
    out[i] = b;
}

// ---------------------------------------------------------------------------
// Stage 3: out[dst, :] += T[src, :]  via hardware fp32 atomics.
// One wave per edge pair; each lane owns 2 of the 64 columns.
// T and out are L2-resident (128 MB each vs 192 MB L2).
// ---------------------------------------------------------------------------
__global__ __launch_bounds__(256) void scatter_add_kernel(
    const int2* __restrict__ nbr,     // [P] (dst, src)
    const float2* __restrict__ T,     // [E, 32] as float2
    float* __restrict__ out,          // [E, 64]
    int nPairs)
{
    const int wave = blockIdx.x * (blockDim.x >> 5) + (threadIdx.x >> 5);
    if (wave >= nPairs) return;
    const int lane = threadIdx.x & 31;

    const int2 e = nbr[wave];                       // x = dst, y = src
    const float2 v = T[(size_t)e.y * (U_DIM / 2) + lane];
    float* __restrict__ o = out + (size_t)e.x * U_DIM + lane * 2;

    unsafeAtomicAdd(o + 0, v.x);   // -> global_atomic_add_f32 (no return)
    unsafeAtomicAdd(o + 1, v.y);
}

// ---------------------------------------------------------------------------
extern "C" void kernel_launch(void* const* d_in, const int* in_sizes, int n_in,
                              void* d_out, int out_size, void* d_ws, size_t ws_size,
                              hipStream_t stream)
{
    const float* feats = (const float*)d_in[0];   // [E, 64]
    const int2*  nbr   = (const int2*)d_in[1];    // [P, 2]
    const float* kern  = (const float*)d_in[2];   // [64, 64]
    const float* bias  = (const float*)d_in[3];   // [64]
    float* out = (float*)d_out;                   // [E, 64]
    float* T   = (float*)d_ws;                    // [E, 64] scratch (128 MB)

    // Stage 1: dense GEMM with WMMA (commuted ahead of the segment sum).
    {
        const int waves  = (E_CNT + 15) / 16;         // 31250
        const int blocks = (waves + 7) / 8;           // 8 waves / 256-thread block
        gemm_wmma_kernel<<<blocks, 256, 0, stream>>>(feats, kern, T, E_CNT);
    }

    // Stage 2: out = broadcast(bias).
    {
        const int nQuads = E_CNT * U_DIM / 4;         // 8M float4
        const int blocks = (nQuads + 255) / 256;
        bias_init_kernel<<<blocks, 256, 0, stream>>>(bias, (float4*)out, nQuads);
    }

    // Stage 3: scatter-add transformed rows into out.
    {
        const int waves  = P_CNT;                     // one wave per pair
        const int blocks = (waves + 7) / 8;
        scatter_add_kernel<<<blocks, 256, 0, stream>>>(nbr, (const float2*)T, out, P_CNT);
    }
}